// OnlineTrans_60421599920562
// MI455X (gfx1250) — compile-verified
//
#include <hip/hip_runtime.h>

// Kronecker matmul  y = x @ kron(L,R)  ==  per-token  Y = L^T (X R)
// Token = 64x64 fp32 matrix; 16384 tokens. Two 64^3 fp32 GEMMs per token
// using V_WMMA_F32_16X16X4_F32 (wave32, 16x16 f32 tiles, K-step 4).

typedef __attribute__((ext_vector_type(2))) float v2f;
typedef __attribute__((ext_vector_type(4))) float v4f;
typedef __attribute__((ext_vector_type(8))) float v8f;

#define WAVES 4                  // one token per wave
#define BLOCK (32 * WAVES)

__global__ __launch_bounds__(BLOCK) void kron_wmma_f32_kernel(
    const float* __restrict__ x,     // [ntok, 64, 64] row-major (l, r)
    const float* __restrict__ L,     // [64, 64] row-major (l, i)
    const float* __restrict__ R,     // [64, 64] row-major (r, j)
    float* __restrict__ out,         // [ntok, 64, 64] row-major (i, j)
    int ntok)
{
    // ldsLT[i*64 + l] = L[l][i]   (L^T row-major -> A-fragments are ds_load_b64)
    // ldsR [j*64 + r] = R[r][j]   (R column-major -> B-fragments are ds_load_b64)
    __shared__ float ldsLT[64 * 64];
    __shared__ float ldsR [64 * 64];
    __shared__ float ldsY1[WAVES * 64 * 64];   // per-wave Y1 scratch, column-major

    const int tid = threadIdx.x;

    // Stage constants (L2-resident after first block; tiny vs. 512MB stream).
    for (int idx = tid; idx < 64 * 64; idx += BLOCK) {
        const int a = idx >> 6;       // dst row
        const int b = idx & 63;       // dst col
        ldsLT[idx] = L[b * 64 + a];
        ldsR [idx] = R[b * 64 + a];
    }
    __syncthreads();

    const int w    = tid >> 5;
    const int lane = tid & 31;
    const int half = lane >> 4;       // 0: lanes 0-15, 1: lanes 16-31
    const int lm   = lane & 15;

    const int token = blockIdx.x * WAVES + w;
    if (token >= ntok) return;

    const float* xt = x   + (size_t)token * 4096;
    float*       yt = out + (size_t)token * 4096;
    float*       y1 = &ldsY1[w * 4096];

    // ------------- GEMM 1: Y1 = X * R  (A = X[l][r], B = R[r][j]) -------------
    #pragma unroll
    for (int lb = 0; lb < 4; ++lb) {
        // A-fragments for this 16-row block of X: read each HBM byte exactly once,
        // keep in 32 VGPRs, reuse across the 4 jb tiles.
        v2f xf[16];
        #pragma unroll
        for (int kc = 0; kc < 16; ++kc)
            xf[kc] = *(const v2f*)(xt + (16 * lb + lm) * 64 + 4 * kc + 2 * half);

        #pragma unroll
        for (int jb = 0; jb < 4; ++jb) {
            v8f acc = {};
            #pragma unroll
            for (int kc = 0; kc < 16; ++kc) {
                v2f bf = *(const v2f*)&ldsR[(16 * jb + lm) * 64 + 4 * kc + 2 * half];
                acc = __builtin_amdgcn_wmma_f32_16x16x4_f32(
                          false, xf[kc], false, bf, (short)0, acc, false, false);
            }
            // C/D layout: lane holds M = 8*half + v (v=0..7), N = lm.
            // Column-major Y1 => the 8 values are contiguous: two ds_store_b128.
            float* dst = &y1[(16 * jb + lm) * 64 + 16 * lb + 8 * half];
            v4f lo = {acc[0], acc[1], acc[2], acc[3]};
            v4f hi = {acc[4], acc[5], acc[6], acc[7]};
            *(v4f*)(dst)     = lo;
            *(v4f*)(dst + 4) = hi;
        }
    }
    // Y1 scratch is wave-private; same-wave DS ops are in-order (DScnt).

    // ------------- GEMM 2: Y = L^T * Y1  (A = LT[i][l], B = Y1[l][j]) -------------
    #pragma unroll
    for (int ib = 0; ib < 4; ++ib) {
        v2f lf[16];
        #pragma unroll
        for (int kc = 0; kc < 16; ++kc)
            lf[kc] = *(const v2f*)&ldsLT[(16 * ib + lm) * 64 + 4 * kc + 2 * half];

        #pragma unroll
        for (int jb = 0; jb < 4; ++jb) {
            v8f acc = {};
            #pragma unroll
            for (int kc = 0; kc < 16; ++kc) {
                v2f bf = *(const v2f*)&y1[(16 * jb + lm) * 64 + 4 * kc + 2 * half];
                acc = __builtin_amdgcn_wmma_f32_16x16x4_f32(
                          false, lf[kc], false, bf, (short)0, acc, false, false);
            }
            // Store: row 16*ib + 8*half + v, col 16*jb + lm. Each instruction
            // writes two 64B contiguous segments; L2 write-combines full lines.
            #pragma unroll
            for (int v = 0; v < 8; ++v)
                yt[(16 * ib + 8 * half + v) * 64 + 16 * jb + lm] = acc[v];
        }
    }
}

extern "C" void kernel_launch(void* const* d_in, const int* in_sizes, int n_in,
                              void* d_out, int out_size, void* d_ws, size_t ws_size,
                              hipStream_t stream) {
    const float* x = (const float*)d_in[0];
    const float* L = (const float*)d_in[1];
    const float* R = (const float*)d_in[2];
    float* out = (float*)d_out;

    const int ntok   = in_sizes[0] / 4096;     // B*S tokens of 64x64
    const int blocks = (ntok + WAVES - 1) / WAVES;

    hipLaunchKernelGGL(kron_wmma_f32_kernel, dim3(blocks), dim3(BLOCK), 0, stream,
                       x, L, R, out, ntok);
}